// S4DLayer_867583394236
// MI455X (gfx1250) — compile-verified
//
#include <hip/hip_runtime.h>
#include <hip/hip_bf16.h>

#define D_MODEL 512
#define STATE_N 64
#define BATCH   4
#define SEQ     2048
#define MROWS   (BATCH * SEQ)      // 8192 rows
#define ET      (D_MODEL / 16)     // 32 output-col tiles
#define MT      (MROWS / 16)       // 512 row tiles

typedef __attribute__((ext_vector_type(2))) float v2f;
typedef __attribute__((ext_vector_type(8))) float v8f;

// ---------------------------------------------------------------------------
// f32 WMMA 16x16x4 fragment addressing (ISA 7.12.2):
//   A (16x4, MxK): lane L -> m = L%16 ; VGPR v -> k = v + 2*(L>=16)
//   B (4x16, KxN): lane L -> n = L%16 ; VGPR v -> k = v + 2*(L>=16)
//   C/D (16x16)  : lane L -> n = L%16 ; VGPR v -> m = v + 8*(L>=16)
// einsum 'btd,ed->bte': out[m,e] = sum_d A[m,d] * W[e,d], so the WMMA
// B-operand element (k,n) = W[n*D + k]. For a fixed lane both A and B
// fragment elements are CONTIGUOUS f32 pairs in the row-major sources,
// so fragments load directly with b64 loads -- no packing/transpose pass.
// ---------------------------------------------------------------------------

// Fused projection kernel: u = x @ W_in^T ; g = sigmoid(x @ W_gate^T + b).
// One wave per 16x16 output tile; A fragment shared by both WMMA chains.
__global__ void k_gemm_ug(const float* __restrict__ X,
                          const float* __restrict__ Win,
                          const float* __restrict__ Wgate,
                          const float* __restrict__ b_gate,
                          float* __restrict__ U, float* __restrict__ G) {
  int wave = blockIdx.x * (blockDim.x >> 5) + (threadIdx.x >> 5);
  int lane = threadIdx.x & 31;
  int mt = wave / ET, et = wave % ET;
  int r  = lane & 15;
  int c2 = (lane >> 4) << 1;                 // k sub-offset: 0 or 2
  const float* ap = X     + (size_t)(mt * 16 + r) * D_MODEL + c2;
  const float* bi = Win   + (size_t)(et * 16 + r) * D_MODEL + c2;
  const float* bg = Wgate + (size_t)(et * 16 + r) * D_MODEL + c2;
  v8f accU = {}, accG = {};
#pragma unroll 8
  for (int k = 0; k < D_MODEL; k += 4) {
    v2f a  = *(const v2f*)(ap + k);
    v2f wi = *(const v2f*)(bi + k);
    v2f wg = *(const v2f*)(bg + k);
    accU = __builtin_amdgcn_wmma_f32_16x16x4_f32(
        false, a, false, wi, (short)0, accU, false, false);
    accG = __builtin_amdgcn_wmma_f32_16x16x4_f32(
        false, a, false, wg, (short)0, accG, false, false);
  }
  int mbase = mt * 16 + ((lane >> 4) << 3);
  int n     = et * 16 + r;
  float bgv = b_gate[n];
#pragma unroll
  for (int v = 0; v < 8; ++v) {
    int idx = (mbase + v) * D_MODEL + n;
    U[idx] = accU[v];
    G[idx] = 1.0f / (1.0f + __expf(-(accG[v] + bgv)));
  }
}

// Output kernel: out = x + g * (y @ W_out^T)
__global__ void k_gemm_out(const float* __restrict__ Y,
                           const float* __restrict__ Wout,
                           const float* __restrict__ X,
                           const float* __restrict__ G,
                           float* __restrict__ OUT) {
  int wave = blockIdx.x * (blockDim.x >> 5) + (threadIdx.x >> 5);
  int lane = threadIdx.x & 31;
  int mt = wave / ET, et = wave % ET;
  int r  = lane & 15;
  int c2 = (lane >> 4) << 1;
  const float* ap = Y    + (size_t)(mt * 16 + r) * D_MODEL + c2;
  const float* bp = Wout + (size_t)(et * 16 + r) * D_MODEL + c2;
  v8f acc = {};
#pragma unroll 8
  for (int k = 0; k < D_MODEL; k += 4) {
    v2f a = *(const v2f*)(ap + k);
    v2f b = *(const v2f*)(bp + k);
    acc = __builtin_amdgcn_wmma_f32_16x16x4_f32(
        false, a, false, b, (short)0, acc, false, false);
  }
  int mbase = mt * 16 + ((lane >> 4) << 3);
  int n     = et * 16 + r;
#pragma unroll
  for (int v = 0; v < 8; ++v) {
    int idx  = (mbase + v) * D_MODEL + n;
    OUT[idx] = X[idx] + G[idx] * acc[v];
  }
}

// ---------------------------------------------------------------------------
// Diagonal-SSM recurrence scan. 4 threads per (b, d) row; each thread owns
// 16 of the 64 complex states in registers. h[t] = r*h[t-1] + u[t] with
// r = exp(dt*A); output weight w_n = C_n * B_n * dt (real, since B_p/C_p
// are real). Cross-lane reduce (groups of 4 lanes) via shfl_xor.
// y[b,t,d] = sum_n w_n * Re(h_n[t]) + D_p[d] * u[b,t,d]
// ---------------------------------------------------------------------------
__global__ void k_scan(const float* __restrict__ U,
                       const float* __restrict__ log_A_real,
                       const float* __restrict__ A_imag,
                       const float* __restrict__ B_p,
                       const float* __restrict__ C_p,
                       const float* __restrict__ D_p,
                       const float* __restrict__ log_dt,
                       float* __restrict__ Y) {
  int gid = blockIdx.x * blockDim.x + threadIdx.x;   // 8192 threads
  int sub = gid & 3;
  int row = gid >> 2;                                // 0..B*D-1
  int d   = row & (D_MODEL - 1);
  int b   = row >> 9;                                // row / 512
  float dt = __expf(log_dt[d]);

  const int NS = 16;
  float hr[NS], hi_[NS], rr[NS], ri[NS], w[NS];
  int nb = sub * NS;
#pragma unroll
  for (int i = 0; i < NS; ++i) {
    int n     = d * STATE_N + nb + i;
    float are = -__expf(log_A_real[n]);
    float aim = A_imag[n];
    float mag = __expf(dt * are);
    float s, c;
    __sincosf(dt * aim, &s, &c);
    rr[i] = mag * c;
    ri[i] = mag * s;
    w[i]  = C_p[n] * B_p[n] * dt;
    hr[i] = 0.0f;
    hi_[i] = 0.0f;
  }
  float dp = D_p[d];
  const float* up = U + (size_t)b * SEQ * D_MODEL + d;
  float*       yp = Y + (size_t)b * SEQ * D_MODEL + d;

  for (int t = 0; t < SEQ; ++t) {
    float ut = up[t * D_MODEL];
    float p  = 0.0f;
#pragma unroll
    for (int i = 0; i < NS; ++i) {
      float nhr = rr[i] * hr[i] - ri[i] * hi_[i] + ut;
      float nhi = ri[i] * hr[i] + rr[i] * hi_[i];
      hr[i] = nhr;
      hi_[i] = nhi;
      p = fmaf(w[i], nhr, p);
    }
    p += __shfl_xor(p, 1, 32);
    p += __shfl_xor(p, 2, 32);
    if (sub == 0)
      yp[t * D_MODEL] = p + dp * ut;
  }
}

// ---------------------------------------------------------------------------
// Launcher. Inputs (setup_inputs order):
//  0 x [B,T,D] 1 log_A_real [D,N] 2 A_imag [D,N] 3 B_p [D,N] 4 C_p [D,N]
//  5 D_p [D]   6 log_dt [D]       7 W_in [D,D]   8 W_out [D,D] 9 W_gate [D,D]
// 10 b_gate [D]
// ---------------------------------------------------------------------------
extern "C" void kernel_launch(void* const* d_in, const int* in_sizes, int n_in,
                              void* d_out, int out_size, void* d_ws, size_t ws_size,
                              hipStream_t stream) {
  (void)in_sizes; (void)n_in; (void)out_size; (void)ws_size;
  const float* x          = (const float*)d_in[0];
  const float* log_A_real = (const float*)d_in[1];
  const float* A_imag     = (const float*)d_in[2];
  const float* B_p        = (const float*)d_in[3];
  const float* C_p        = (const float*)d_in[4];
  const float* D_p        = (const float*)d_in[5];
  const float* log_dt     = (const float*)d_in[6];
  const float* W_in       = (const float*)d_in[7];
  const float* W_out      = (const float*)d_in[8];
  const float* W_gate     = (const float*)d_in[9];
  const float* b_gate     = (const float*)d_in[10];
  float* out = (float*)d_out;

  // Workspace: three f32 [M, D] buffers (48 MB total)
  float* u = (float*)d_ws;
  float* g = u + (size_t)MROWS * D_MODEL;
  float* y = g + (size_t)MROWS * D_MODEL;

  const int TPB = 256;
  dim3 blk(TPB);
  const int GEMM_BLOCKS = (MT * ET) / (TPB / 32);   // 16384 waves, 8/block

  // 1) u = x @ W_in^T ; g = sigmoid(x @ W_gate^T + b_gate)   (fused, f32 WMMA)
  k_gemm_ug<<<dim3(GEMM_BLOCKS), blk, 0, stream>>>(x, W_in, W_gate, b_gate, u, g);
  // 2) SSM recurrence scan -> y (includes D_p*u skip term)
  k_scan<<<dim3((BATCH * D_MODEL * 4) / TPB), blk, 0, stream>>>(
      u, log_A_real, A_imag, B_p, C_p, D_p, log_dt, y);
  // 3) out = x + g * (y @ W_out^T)
  k_gemm_out<<<dim3(GEMM_BLOCKS), blk, 0, stream>>>(y, W_out, x, g, out);
}